// DGCNNConv_87376814670237
// MI455X (gfx1250) — compile-verified
//
#include <hip/hip_runtime.h>

typedef __attribute__((ext_vector_type(2))) float v2f;
typedef __attribute__((ext_vector_type(8))) float v8f;

#define NB 16
#define NP 2048
#define KNN 20
#define NEG_BIG (-3.402823466e38f)

// ---------------------------------------------------------------------------
// x [B,3,N] -> xt [B,N,3]
// ---------------------------------------------------------------------------
__global__ void transpose_x_kernel(const float* __restrict__ x, float* __restrict__ xt) {
  int t = blockIdx.x * blockDim.x + threadIdx.x;
  if (t >= NB * 3 * NP) return;
  int b = t / (3 * NP);
  int rem = t - b * 3 * NP;
  int c = rem / NP;
  int n = rem - c * NP;
  xt[((size_t)b * NP + n) * 3 + c] = x[t];
}

// ---------------------------------------------------------------------------
// sq[r] = ||F[r,:]||^2 for all B*N rows
// ---------------------------------------------------------------------------
__global__ void rowsq_kernel(const float* __restrict__ F, int fs, int C,
                             float* __restrict__ sq) {
  int r = blockIdx.x * blockDim.x + threadIdx.x;
  if (r >= NB * NP) return;
  const float* p = F + (size_t)r * fs;
  float s = 0.f;
  for (int c = 0; c < C; ++c) s += p[c] * p[c];
  sq[r] = s;
}

// ---------------------------------------------------------------------------
// Per-batch negative squared distance matrix via f32 WMMA:
//   dist[m,n] = 2*F[m].F[n] - sq[m] - sq[n]
// One 128-thread block computes a 64x64 output block: row strip + col strip
// staged (zero-padded to Cp) in LDS, each wave computes 4 16x16 tiles with a
// shared A fragment. Inner loop: 5x ds_load_b64 + 4x v_wmma, no guards.
// ---------------------------------------------------------------------------
__global__ void gram_kernel(const float* __restrict__ F, int fs, int C, int Cp,
                            const float* __restrict__ sq, float* __restrict__ dist) {
  extern __shared__ float sm[];
  int wave = threadIdx.x >> 5, lane = threadIdx.x & 31;
  int half = lane >> 4, lid = lane & 15;
  int pitch = Cp + 4;
  int row0 = blockIdx.y << 6, col0 = blockIdx.x << 6;
  float* rS = sm;
  float* cS = sm + (size_t)64 * pitch;

  // stage 64 row-rows + 64 col-rows, zero-padded to Cp
  for (int s = wave; s < 128; s += 4) {
    int isCol = s >> 6;
    int r = s & 63;
    const float* src = F + (size_t)((isCol ? col0 : row0) + r) * fs;
    float* dst = (isCol ? cS : rS) + (size_t)r * pitch;
    for (int c = lane; c < C; c += 32) dst[c] = src[c];
    for (int c = C + lane; c < Cp; c += 32) dst[c] = 0.f;
  }
  __syncthreads();

  const float* aRow = rS + (size_t)(wave * 16 + lid) * pitch;
  v8f acc[4] = {};
  for (int k0 = 0; k0 < Cp; k0 += 4) {
    int ka = k0 + 2 * half;
    v2f a = *(const v2f*)&aRow[ka];
#pragma unroll
    for (int t = 0; t < 4; ++t) {
      v2f b = *(const v2f*)&cS[(size_t)(t * 16 + lid) * pitch + ka];
      acc[t] = __builtin_amdgcn_wmma_f32_16x16x4_f32(false, a, false, b, (short)0,
                                                     acc[t], false, false);
    }
  }

  int mbase = row0 + wave * 16 + 8 * half;
  float sqm[8];
#pragma unroll
  for (int r = 0; r < 8; ++r) sqm[r] = sq[mbase + r];
#pragma unroll
  for (int t = 0; t < 4; ++t) {
    int col = col0 + t * 16 + lid;
    float sqc = sq[col];
#pragma unroll
    for (int r = 0; r < 8; ++r) {
      dist[(size_t)(mbase + r) * NP + col] = 2.f * acc[t][r] - sqm[r] - sqc;
    }
  }
}

// ---------------------------------------------------------------------------
// Per-batch top-20 (largest) per row of dist -> nidx [N,20].
// One thread per row; dist is L2-resident (16MB << 192MB L2). float4 loads,
// branchless register insertion list.
// ---------------------------------------------------------------------------
__global__ void topk_kernel(const float* __restrict__ dist, int* __restrict__ nidx) {
  int row = blockIdx.x * blockDim.x + threadIdx.x;  // 8 * 256 = 2048 rows
  const float4* d4 = (const float4*)(dist + (size_t)row * NP);
  float v[KNN];
  int id[KNN];
#pragma unroll
  for (int i = 0; i < KNN; ++i) { v[i] = NEG_BIG; id[i] = 0; }
  for (int j = 0; j < NP / 4; ++j) {
    float4 x4 = d4[j];
    float xs[4] = {x4.x, x4.y, x4.z, x4.w};
#pragma unroll
    for (int u = 0; u < 4; ++u) {
      float x = xs[u];
      if (x > v[KNN - 1]) {
        float cv = x;
        int ci = 4 * j + u;
#pragma unroll
        for (int q = 0; q < KNN; ++q) {
          bool sw = cv > v[q];
          float tv = v[q]; int ti = id[q];
          v[q]  = sw ? cv : tv;
          id[q] = sw ? ci : ti;
          cv = sw ? tv : cv;
          ci = sw ? ti : ci;
        }
      }
    }
  }
  int* o = nidx + row * KNN;
#pragma unroll
  for (int i = 0; i < KNN; ++i) o[i] = id[i];
}

// ---------------------------------------------------------------------------
// Fused gather + 1x1-conv GEMM via f32 WMMA. Two modes:
//   STATS=true  : accumulate per-channel sum / sumsq of h (BN pass A)
//   STATS=false : apply BN scale/shift + LeakyReLU and either
//                 - edge mode (nidx != null): segmented max over k=20 via atomicMax
//                 - fc mode   (nidx == null): direct transposed store to out[B,O,N]
// 4 waves/block; each wave owns one 16-row edge tile in LDS (zero-padded to P);
// the 16xP W tile for the current output tile is staged cooperatively in LDS.
// Inner K-loop: 2x ds_load_b64 + 1x v_wmma, guard-free.
// ---------------------------------------------------------------------------
template <bool STATS>
__global__ void conv_kernel(const float* __restrict__ F, int fs, int Cin,
                            const int* __restrict__ nidx,
                            const float* __restrict__ W, int twoC, int P, int O,
                            const float* __restrict__ scale,
                            const float* __restrict__ shift,
                            float* __restrict__ outp, int ostride,
                            float* __restrict__ gsum, float* __restrict__ gsq) {
  extern __shared__ float sm[];
  const int WAVES = 4;
  int wave = threadIdx.x >> 5, lane = threadIdx.x & 31;
  int half = lane >> 4, lid = lane & 15;
  int pitch = P + 4;  // dword stride mod 64 banks = 4*odd -> conflict-free b64
  float* eS = sm + (size_t)wave * 16 * pitch;
  float* wS = sm + (size_t)WAVES * 16 * pitch;
  float* sSum = wS + (size_t)16 * pitch;  // [O] (STATS only)
  float* sSq = sSum + O;
  if (STATS) {
    for (int t = threadIdx.x; t < 2 * O; t += blockDim.x) sSum[t] = 0.f;
  }

  int tile0 = (blockIdx.x * WAVES + wave) * 16;

  // ---- stage edge-feature tile [16 x P] into LDS (zero-padded, guard-free) ----
  for (int r = 0; r < 16; ++r) {
    int g = tile0 + r;
    float* erow = eS + (size_t)r * pitch;
    if (nidx) {
      int bn = g / KNN;  // b*N + n
      int j = nidx[g];
      int b = bn >> 11;  // N = 2048
      const float* ctr = F + (size_t)bn * fs;
      const float* nbr = F + ((size_t)(b << 11) + j) * fs;
      for (int c = lane; c < Cin; c += 32) erow[c] = nbr[c] - ctr[c];
      for (int c = Cin + lane; c < twoC; c += 32) erow[c] = ctr[c - Cin];
    } else {
      const float* ctr = F + (size_t)g * fs;
      for (int c = lane; c < twoC; c += 32) erow[c] = ctr[c];
    }
    for (int c = twoC + lane; c < P; c += 32) erow[c] = 0.f;
  }

  for (int ob = 0; ob < O; ob += 16) {
    __syncthreads();  // protect wS from previous iteration's readers
    for (int r = wave; r < 16; r += WAVES) {
      const float* wrow = W + (size_t)(ob + r) * twoC;
      float* wd = wS + (size_t)r * pitch;
      for (int c = lane; c < twoC; c += 32) wd[c] = wrow[c];
      for (int c = twoC + lane; c < P; c += 32) wd[c] = 0.f;
    }
    __syncthreads();

    v8f acc = {};
    const float* arow = eS + (size_t)lid * pitch;
    const float* brow = wS + (size_t)lid * pitch;
    for (int k0 = 0; k0 < P; k0 += 4) {
      int ka = k0 + 2 * half;
      v2f a = *(const v2f*)&arow[ka];
      v2f b = *(const v2f*)&brow[ka];
      acc = __builtin_amdgcn_wmma_f32_16x16x4_f32(false, a, false, b, (short)0, acc,
                                                  false, false);
    }

    if (STATS) {
      float s = 0.f, q = 0.f;
#pragma unroll
      for (int r = 0; r < 8; ++r) { float v = acc[r]; s += v; q += v * v; }
      // lanes l and l^16 hold the same output channel (N = lane&15)
      s += __shfl_xor(s, 16, 32);
      q += __shfl_xor(q, 16, 32);
      if (half == 0) {
        atomicAdd(&sSum[ob + lid], s);
        atomicAdd(&sSq[ob + lid], q);
      }
    } else {
      float sc = scale[ob + lid], sh = shift[ob + lid];
      int gbase = tile0 + 8 * half;
      if (nidx) {
        // segmented max over neighbor groups (g/KNN), flush via float atomicMax
        int curg = -1;
        float run = 0.f;
#pragma unroll
        for (int r = 0; r < 8; ++r) {
          int g = gbase + r;
          int grp = g / KNN;
          float v = acc[r] * sc + sh;
          v = v > 0.f ? v : 0.2f * v;
          if (grp != curg) {
            if (curg >= 0) atomicMax(&outp[(size_t)curg * ostride + ob + lid], run);
            curg = grp;
            run = v;
          } else {
            run = fmaxf(run, v);
          }
        }
        atomicMax(&outp[(size_t)curg * ostride + ob + lid], run);
      } else {
        // fc mode: out[b][o][n], 8 consecutive n per lane -> two b128 stores
        float vv[8];
#pragma unroll
        for (int r = 0; r < 8; ++r) {
          float v = acc[r] * sc + sh;
          vv[r] = v > 0.f ? v : 0.2f * v;
        }
        int b = gbase >> 11;
        int n0 = gbase & (NP - 1);
        float* dst = outp + ((size_t)b * O + ob + lid) * NP + n0;
        *(float4*)(dst)     = make_float4(vv[0], vv[1], vv[2], vv[3]);
        *(float4*)(dst + 4) = make_float4(vv[4], vv[5], vv[6], vv[7]);
      }
    }
  }

  if (STATS) {
    __syncthreads();
    for (int t = threadIdx.x; t < O; t += blockDim.x) {
      atomicAdd(&gsum[t], sSum[t]);
      atomicAdd(&gsq[t], sSq[t]);
    }
  }
}

// ---------------------------------------------------------------------------
// BN finalize: scale = g * rsqrt(var + eps), shift = b - mean * scale
// ---------------------------------------------------------------------------
__global__ void bn_finalize_kernel(const float* __restrict__ sum,
                                   const float* __restrict__ sumsq,
                                   const float* __restrict__ g,
                                   const float* __restrict__ b, float invR, int O,
                                   float* __restrict__ scale,
                                   float* __restrict__ shift) {
  int o = blockIdx.x * blockDim.x + threadIdx.x;
  if (o >= O) return;
  float m = sum[o] * invR;
  float var = sumsq[o] * invR - m * m;
  float sc = g[o] * rsqrtf(var + 1e-5f);
  scale[o] = sc;
  shift[o] = b[o] - m * sc;
}

__global__ void fill_kernel(float* __restrict__ p, float v, int n) {
  int t = blockIdx.x * blockDim.x + threadIdx.x;
  if (t < n) p[t] = v;
}

__global__ void fill_slice_kernel(float* __restrict__ base, int cols, int stride,
                                  float v, int total) {
  int t = blockIdx.x * blockDim.x + threadIdx.x;
  if (t >= total) return;
  base[(size_t)(t / cols) * stride + (t % cols)] = v;
}

// ---------------------------------------------------------------------------
// Host orchestration
// ---------------------------------------------------------------------------
extern "C" void kernel_launch(void* const* d_in, const int* in_sizes, int n_in,
                              void* d_out, int out_size, void* d_ws, size_t ws_size,
                              hipStream_t stream) {
  const float* x = (const float*)d_in[0];
  const float* w[5]  = {(const float*)d_in[1], (const float*)d_in[4],
                        (const float*)d_in[7], (const float*)d_in[10],
                        (const float*)d_in[13]};
  const float* gg[5] = {(const float*)d_in[2], (const float*)d_in[5],
                        (const float*)d_in[8], (const float*)d_in[11],
                        (const float*)d_in[14]};
  const float* bb[5] = {(const float*)d_in[3], (const float*)d_in[6],
                        (const float*)d_in[9], (const float*)d_in[12],
                        (const float*)d_in[15]};
  float* out = (float*)d_out;

  // workspace layout (floats)
  float* feat = (float*)d_ws;                  // [B*N, 512] = x1|x2|x3|x4 slices
  float* xt   = feat + (size_t)NB * NP * 512;  // [B*N, 3]
  float* sq   = xt + (size_t)NB * NP * 3;      // [B*N]
  float* dist = sq + (size_t)NB * NP;          // [N, N] (per-batch reuse)
  float* ssum = dist + (size_t)NP * NP;        // [512]
  float* ssq  = ssum + 512;                    // [512]
  float* scl  = ssq + 512;                     // [512]
  float* sft  = scl + 512;                     // [512]
  int* nidx   = (int*)(sft + 512);             // [B*N, 20]

  const int R_EDGE = NB * NP * KNN;  // 655360, multiple of 64
  const int R_FC = NB * NP;          // 32768,  multiple of 64

  transpose_x_kernel<<<(NB * 3 * NP + 255) / 256, 256, 0, stream>>>(x, xt);

  struct LayerCfg { const float* Fin; int fs; int Cin; float* outSlice; int O; };
  LayerCfg L[4] = {
      {xt, 3, 3, feat, 64},
      {feat, 512, 64, feat + 64, 64},
      {feat + 64, 512, 64, feat + 128, 128},
      {feat + 128, 512, 128, feat + 256, 256},
  };

  for (int l = 0; l < 4; ++l) {
    const LayerCfg& c = L[l];
    // --- KNN: per-batch Gram (WMMA, 64x64 blocks) + top-20 ---
    rowsq_kernel<<<(NB * NP + 255) / 256, 256, 0, stream>>>(c.Fin, c.fs, c.Cin, sq);
    int Cp = (c.Cin + 3) & ~3;
    size_t shG = (size_t)128 * (Cp + 4) * sizeof(float);
    for (int b = 0; b < NB; ++b) {
      gram_kernel<<<dim3(NP / 64, NP / 64), 128, shG, stream>>>(
          c.Fin + (size_t)b * NP * c.fs, c.fs, c.Cin, Cp, sq + (size_t)b * NP, dist);
      topk_kernel<<<NP / 256, 256, 0, stream>>>(dist, nidx + (size_t)b * NP * KNN);
    }
    // --- BN pass A: WMMA GEMM + channel stats ---
    int twoC = 2 * c.Cin;
    int P = (twoC + 7) & ~7;
    size_t shA = (size_t)(4 * 16 + 16) * (P + 4) * sizeof(float) +
                 (size_t)2 * c.O * sizeof(float);
    size_t shB = (size_t)(4 * 16 + 16) * (P + 4) * sizeof(float);
    fill_kernel<<<(1024 + 255) / 256, 256, 0, stream>>>(ssum, 0.f, 1024);  // ssum+ssq
    conv_kernel<true><<<R_EDGE / 64, 128, shA, stream>>>(
        c.Fin, c.fs, c.Cin, nidx, w[l], twoC, P, c.O, nullptr, nullptr, nullptr, 0,
        ssum, ssq);
    bn_finalize_kernel<<<(c.O + 63) / 64, 64, 0, stream>>>(
        ssum, ssq, gg[l], bb[l], 1.f / (float)R_EDGE, c.O, scl, sft);
    // --- BN pass B: recompute GEMM, bn+lrelu, max over k via atomicMax ---
    fill_slice_kernel<<<(NB * NP * c.O + 255) / 256, 256, 0, stream>>>(
        c.outSlice, c.O, 512, NEG_BIG, NB * NP * c.O);
    conv_kernel<false><<<R_EDGE / 64, 128, shB, stream>>>(
        c.Fin, c.fs, c.Cin, nidx, w[l], twoC, P, c.O, scl, sft, c.outSlice, 512,
        nullptr, nullptr);
  }

  // --- layer 5: fc conv over concat features, BN over (B,N), transposed out ---
  {
    const int O = 512, twoC = 512, P = 512;
    size_t shA = (size_t)(4 * 16 + 16) * (P + 4) * sizeof(float) +
                 (size_t)2 * O * sizeof(float);
    size_t shB = (size_t)(4 * 16 + 16) * (P + 4) * sizeof(float);
    fill_kernel<<<(1024 + 255) / 256, 256, 0, stream>>>(ssum, 0.f, 1024);
    conv_kernel<true><<<R_FC / 64, 128, shA, stream>>>(
        feat, 512, 512, nullptr, w[4], twoC, P, O, nullptr, nullptr, nullptr, 0,
        ssum, ssq);
    bn_finalize_kernel<<<(O + 63) / 64, 64, 0, stream>>>(
        ssum, ssq, gg[4], bb[4], 1.f / (float)R_FC, O, scl, sft);
    conv_kernel<false><<<R_FC / 64, 128, shB, stream>>>(
        feat, 512, 512, nullptr, w[4], twoC, P, O, scl, sft, out, 0, nullptr,
        nullptr);
  }
}